// GraphSAGE_12008728560245
// MI455X (gfx1250) — compile-verified
//
#include <hip/hip_runtime.h>
#include <hip/hip_bf16.h>

// ---------------------------------------------------------------------------
// GraphSAGE layer on MI455X (gfx1250, wave32):
//   1) zero workspace (neigh_sum [N,128] + deg [N])
//   2) per-edge scatter: 1 wave per edge, float4 load + 4 f32 L2 atomics/lane
//   3) fused GEMM: out = relu(X@Ws^T + (neigh_sum/deg)@Wn^T + bs + bn)
//      using V_WMMA_F32_16X16X4_F32 (fp32-exact matrix pipe)
// ---------------------------------------------------------------------------

typedef __attribute__((ext_vector_type(2))) float v2f;
typedef __attribute__((ext_vector_type(8))) float v8f;

#define DIM 128

__global__ __launch_bounds__(256) void sage_zero(float* __restrict__ ws, long total) {
    long i = (long)blockIdx.x * blockDim.x + threadIdx.x;
    long stride = (long)gridDim.x * blockDim.x;
    for (; i < total; i += stride) ws[i] = 0.0f;
}

// One wave (32 lanes) per edge: lane l handles feature chunk [4l, 4l+4).
__global__ __launch_bounds__(256) void sage_scatter(const float* __restrict__ feats,
                                                    const int* __restrict__ esrc,
                                                    const int* __restrict__ edst,
                                                    float* __restrict__ neigh_sum,
                                                    float* __restrict__ deg,
                                                    int E) {
    const int lane = threadIdx.x & 31;
    const int edge = blockIdx.x * (blockDim.x >> 5) + (threadIdx.x >> 5);
    if (edge >= E) return;
    const int s = esrc[edge];
    const int d = edst[edge];
    const float4 v = *(const float4*)(feats + (size_t)s * DIM + lane * 4);
    float* p = neigh_sum + (size_t)d * DIM + lane * 4;
    atomicAdd(p + 0, v.x);   // -> global_atomic_add_f32, RMW at L2 (data resident)
    atomicAdd(p + 1, v.y);
    atomicAdd(p + 2, v.z);
    atomicAdd(p + 3, v.w);
    if (lane == 0) atomicAdd(deg + d, 1.0f);
}

// Block = 256 threads = 8 waves. Block owns 16 output rows; wave w owns
// output columns [16w, 16w+16). Accumulates both GEMMs into one v8f C operand.
__global__ __launch_bounds__(256) void sage_gemm(const float* __restrict__ X,
                                                 const float* __restrict__ NS,
                                                 const float* __restrict__ deg,
                                                 const float* __restrict__ Wself,
                                                 const float* __restrict__ bself,
                                                 const float* __restrict__ Wneigh,
                                                 const float* __restrict__ bneigh,
                                                 float* __restrict__ out,
                                                 int N) {
    const int lane = threadIdx.x & 31;
    const int wave = threadIdx.x >> 5;          // column tile 0..7
    const int m0   = blockIdx.x * 16;
    if (m0 >= N) return;

    const int half = lane >> 4;                 // lane half selects K pair
    const int l16  = lane & 15;
    const int koff = half * 2;                  // K = {0,1} or {2,3} within step
    int rowA = m0 + l16;                        // A-fragment row (per ISA layout)
    if (rowA >= N) rowA = N - 1;                // clamp loads (EXEC must stay all-1s)
    const int colB = wave * 16 + l16;           // B-fragment col == output col

    const float* xrow  = X  + (size_t)rowA * DIM + koff;
    const float* nrow  = NS + (size_t)rowA * DIM + koff;
    const float* wsrow = Wself  + (size_t)colB * DIM + koff;
    const float* wnrow = Wneigh + (size_t)colB * DIM + koff;
    const float invd   = 1.0f / fmaxf(deg[rowA], 1.0f);

    v8f c = {};
    // out += X @ Wself^T  : both operands K-contiguous -> b64 loads
#pragma unroll 4
    for (int kk = 0; kk < DIM; kk += 4) {
        const v2f a = *(const v2f*)(xrow + kk);
        const v2f b = *(const v2f*)(wsrow + kk);
        c = __builtin_amdgcn_wmma_f32_16x16x4_f32(
                /*neg_a=*/false, a, /*neg_b=*/false, b,
                /*c_mod=*/(short)0, c, /*reuse_a=*/false, /*reuse_b=*/false);
    }
    // out += (neigh_sum * invd) @ Wneigh^T  (chained through C operand)
#pragma unroll 4
    for (int kk = 0; kk < DIM; kk += 4) {
        v2f a = *(const v2f*)(nrow + kk);
        a.x *= invd;
        a.y *= invd;
        const v2f b = *(const v2f*)(wnrow + kk);
        c = __builtin_amdgcn_wmma_f32_16x16x4_f32(
                false, a, false, b, (short)0, c, false, false);
    }

    const float bias = bself[colB] + bneigh[colB];
#pragma unroll
    for (int v = 0; v < 8; ++v) {
        const int row = m0 + v + half * 8;      // C/D layout: M = vgpr + 8*half
        if (row < N) {
            float val = c[v] + bias;
            out[(size_t)row * DIM + colB] = val > 0.0f ? val : 0.0f;
        }
    }
}

extern "C" void kernel_launch(void* const* d_in, const int* in_sizes, int n_in,
                              void* d_out, int out_size, void* d_ws, size_t ws_size,
                              hipStream_t stream) {
    const float* node_feats = (const float*)d_in[0];
    const int*   edge_src   = (const int*)d_in[1];
    const int*   edge_dst   = (const int*)d_in[2];
    const float* W_self     = (const float*)d_in[3];
    const float* b_self     = (const float*)d_in[4];
    const float* W_neigh    = (const float*)d_in[5];
    const float* b_neigh    = (const float*)d_in[6];
    float*       out        = (float*)d_out;

    const int N = in_sizes[0] / DIM;
    const int E = in_sizes[1];

    float* neigh_sum = (float*)d_ws;                  // N*128 f32
    float* deg       = neigh_sum + (size_t)N * DIM;   // N f32

    // 1) zero neigh_sum + deg
    const long zero_total = (long)N * DIM + N;
    sage_zero<<<2048, 256, 0, stream>>>((float*)d_ws, zero_total);

    // 2) per-edge scatter: 8 waves per 256-thread block -> 8 edges per block
    const int edge_blocks = (E + 7) / 8;
    sage_scatter<<<edge_blocks, 256, 0, stream>>>(node_feats, edge_src, edge_dst,
                                                  neigh_sum, deg, E);

    // 3) fused WMMA GEMM + mean-normalize + bias + relu
    const int row_tiles = (N + 15) / 16;
    sage_gemm<<<row_tiles, 256, 0, stream>>>(node_feats, neigh_sum, deg,
                                             W_self, b_self, W_neigh, b_neigh,
                                             out, N);
}